// LocalWLGNN_90245852823902
// MI455X (gfx1250) — compile-verified
//
#include <hip/hip_runtime.h>
#include <hip/hip_bf16.h>

// ---------------------------------------------------------------------------
// LocalWLGNN 3-hop message passing, MI455X (gfx1250, wave32).
// Bandwidth/atomic-bound scatter-gather: no GEMM structure -> no WMMA.
// CDNA5-specific paths used:
//   * global_load_async_to_lds_b128 + s_wait_asynccnt (async copy engine)
//     to double-buffer per-edge row gathers in LDS, with a one-deep rotating
//     register prefetch of the edge indices feeding it.
//   * wave32 layout: 1 wave == 1 row of D=128 floats (32 lanes x float4).
//   * global_atomic_add_f32 scatter; x/h ping/pong (~154MB) kept L2-resident
//     (192MB L2); streaming `out` accumulator uses TH=NT to avoid evicting
//     the gather working set.
// ---------------------------------------------------------------------------

#define D               128
#define F4_PER_ROW      32          // 128 floats / 4
#define WAVES_PER_BLOCK 8
#define EDGE_BLOCK      256
#define HOPS            3

typedef float v4f __attribute__((ext_vector_type(4)));   // native clang vector

// ---- CDNA5 async global->LDS helpers (inline asm: portable across both
// ---- toolchains; builtin arity differs between ROCm 7.2 and clang-23).
__device__ __forceinline__ void async_ld_b128(unsigned lds_byte_addr, const void* gaddr) {
    // GV mode: per-lane 64-bit global address, per-lane 32-bit LDS address.
    asm volatile("global_load_async_to_lds_b128 %0, %1, off"
                 :: "v"(lds_byte_addr), "v"((unsigned long long)gaddr)
                 : "memory");
}
__device__ __forceinline__ void wait_async_le2() {
    asm volatile("s_wait_asynccnt 0x2" ::: "memory");
}
__device__ __forceinline__ void wait_async_le0() {
    asm volatile("s_wait_asynccnt 0x0" ::: "memory");
}

// ---------------------------------------------------------------------------
// Edge kernel: one wave per edge. For edge e:
//   v = (1+b1)*hsrc[si[e]] + x[ni[e]]        (gathered rows, LDS-staged)
//   sums[ni[e]] += v  (atomics)              touched[ni[e]] = 1
// Pipeline (per wave, stride = total_waves):
//   stage -1: indices of edge e+2*stride prefetched into registers
//   stage  0: async row loads of edge e+stride issued into LDS buf^1
//   stage  1: compute + atomic scatter of edge e from LDS buf
// ---------------------------------------------------------------------------
__global__ void __launch_bounds__(EDGE_BLOCK)
edge_scatter_kernel(const float* __restrict__ x, const float* __restrict__ hsrc,
                    const int* __restrict__ si, const int* __restrict__ ni,
                    const float* __restrict__ beta1, int hop,
                    float* __restrict__ sums, int* __restrict__ touched,
                    int E, int total_waves)
{
    // [wave][buf(2)][row(2: h,x)][D floats]  == 8 * 2KB = 16KB LDS
    __shared__ __align__(16) float lds[WAVES_PER_BLOCK * 2 * 2 * D];

    const int lane  = threadIdx.x & 31;
    const int wave  = threadIdx.x >> 5;
    const int gwave = blockIdx.x * WAVES_PER_BLOCK + wave;
    const float a1  = 1.0f + beta1[hop];

    float* wlds = &lds[wave * (2 * 2 * D)];
    const unsigned wlds_addr = (unsigned)(unsigned long long)wlds; // LDS byte addr (low 32 bits of flat)
    const unsigned lane16 = (unsigned)(lane * 16);

    int e = gwave;
    if (e >= E) return;

    // Prologue: current edge indices + async row loads into buf 0.
    int cur_n;                       // scatter target of the edge being computed
    {
        const int s0 = si[e];
        cur_n        = ni[e];
        async_ld_b128(wlds_addr + lane16,                 hsrc + (size_t)s0 * D + lane * 4);
        async_ld_b128(wlds_addr + (unsigned)(D * 4) + lane16, x + (size_t)cur_n * D + lane * 4);
    }
    // Prefetch indices of the next edge into registers.
    int nxt_s = 0, nxt_n = 0;
    {
        const int en = e + total_waves;
        if (en < E) { nxt_s = si[en]; nxt_n = ni[en]; }
    }

    int buf = 0;
    for (; e < E; e += total_waves) {
        const int en = e + total_waves;
        const int this_n = cur_n;

        if (en < E) {
            // Issue async row loads for the next edge (indices already in regs).
            const unsigned nb = wlds_addr + (unsigned)((buf ^ 1) * 2 * D * 4);
            async_ld_b128(nb + lane16,                   hsrc + (size_t)nxt_s * D + lane * 4);
            async_ld_b128(nb + (unsigned)(D * 4) + lane16, x + (size_t)nxt_n * D + lane * 4);
            cur_n = nxt_n;                       // rotate: en becomes current next iter
            const int en2 = en + total_waves;    // prefetch indices two edges ahead
            if (en2 < E) { nxt_s = si[en2]; nxt_n = ni[en2]; }
            wait_async_le2();   // 2 newest outstanding -> current buffer landed
        } else {
            wait_async_le0();   // drain: current buffer landed
        }

        const float* hb = wlds + buf * 2 * D;        // gathered hsrc row
        const float* xb = hb + D;                    // gathered x row
        const v4f hv = *(const v4f*)(hb + lane * 4); // ds_load_b128
        const v4f xv = *(const v4f*)(xb + lane * 4); // ds_load_b128

        const v4f v = a1 * hv + xv;

        float* dst = sums + (size_t)this_n * D + lane * 4;
        atomicAdd(dst + 0, v.x);                     // global_atomic_add_f32
        atomicAdd(dst + 1, v.y);
        atomicAdd(dst + 2, v.z);
        atomicAdd(dst + 3, v.w);
        if (lane == 0) touched[this_n] = 1;          // benign race: all write 1

        buf ^= 1;
    }
}

// ---------------------------------------------------------------------------
// Node fixup: one thread per float4 of the [N,D] matrix.
//   s = touched ? sums : x
//   h = (1+b3) * (s + 1 + b2*x)          (written back into hbuf -> next hop)
//   out = (hop==0 ? (1+eps)*x : out) + h
// `out` is streaming-only -> non-temporal (TH=NT) so it never evicts the
// L2-resident gather buffers (x, h ping/pong).
// ---------------------------------------------------------------------------
__global__ void __launch_bounds__(256)
node_fixup_kernel(const float* __restrict__ x, float* __restrict__ hbuf,
                  const int* __restrict__ touched, float* __restrict__ out,
                  const float* __restrict__ beta2, const float* __restrict__ beta3,
                  const float* __restrict__ eps, int hop, long long nq)
{
    const long long idx = (long long)blockIdx.x * blockDim.x + threadIdx.x; // float4 index
    if (idx >= nq) return;
    const int node = (int)(idx >> 5);     // 32 float4 per row

    const v4f xv = ((const v4f*)x)[idx];
    const bool t = touched[node] != 0;
    const v4f s = t ? ((const v4f*)hbuf)[idx] : xv;

    const float b2 = beta2[hop];
    const float a3 = 1.0f + beta3[hop];

    const v4f h = a3 * (s + 1.0f + b2 * xv);
    ((v4f*)hbuf)[idx] = h;                // becomes hsrc of next hop (keep RT / L2)

    v4f o;
    if (hop == 0) {
        const float a0 = 1.0f + eps[0];
        o = a0 * xv + h;
    } else {
        o = __builtin_nontemporal_load((const v4f*)out + idx) + h;
    }
    __builtin_nontemporal_store(o, (v4f*)out + idx);
}

// ---------------------------------------------------------------------------
// d_in order: [0]=x [1]=eps [2]=beta1 [3]=beta2 [4]=beta3
//             [5]=si0 [6]=ni0 [7]=si1 [8]=ni1 [9]=si2 [10]=ni2
// d_ws layout: hA[N*D] | hB[N*D] | touched[N]   (~103 MB)
// ---------------------------------------------------------------------------
extern "C" void kernel_launch(void* const* d_in, const int* in_sizes, int n_in,
                              void* d_out, int out_size, void* d_ws, size_t ws_size,
                              hipStream_t stream)
{
    const float* x    = (const float*)d_in[0];
    const float* eps  = (const float*)d_in[1];
    const float* b1   = (const float*)d_in[2];
    const float* b2   = (const float*)d_in[3];
    const float* b3   = (const float*)d_in[4];
    const int* si[HOPS] = {(const int*)d_in[5], (const int*)d_in[7], (const int*)d_in[9]};
    const int* ni[HOPS] = {(const int*)d_in[6], (const int*)d_in[8], (const int*)d_in[10]};

    const int N = in_sizes[0] / D;
    const int E = in_sizes[5];
    float* out = (float*)d_out;

    float* hA      = (float*)d_ws;
    float* hB      = hA + (size_t)N * D;
    int*   touched = (int*)(hB + (size_t)N * D);

    const long long nq = (long long)N * F4_PER_ROW;      // float4 count
    const int fixBlocks = (int)((nq + 255) / 256);

    // ~8 edges per wave -> 2-deep async pipeline amortized over 8 iterations.
    const int waves_needed = (E + 7) / 8;
    const int edgeBlocks   = (waves_needed + WAVES_PER_BLOCK - 1) / WAVES_PER_BLOCK;
    const int total_waves  = edgeBlocks * WAVES_PER_BLOCK;

    const float* hsrc = x;     // hop 0 reads h == x directly (no copy)
    float*       hdst = hA;

    for (int hop = 0; hop < HOPS; ++hop) {
        (void)hipMemsetAsync(hdst, 0, (size_t)N * D * sizeof(float), stream);
        (void)hipMemsetAsync(touched, 0, (size_t)N * sizeof(int), stream);

        edge_scatter_kernel<<<edgeBlocks, EDGE_BLOCK, 0, stream>>>(
            x, hsrc, si[hop], ni[hop], b1, hop, hdst, touched, E, total_waves);

        node_fixup_kernel<<<fixBlocks, 256, 0, stream>>>(
            x, hdst, touched, out, b2, b3, eps, hop, nq);

        hsrc = hdst;
        hdst = (hdst == hA) ? hB : hA;
    }
}